// FlowStateCausalRevIN_2199023255788
// MI455X (gfx1250) — compile-verified
//
#include <hip/hip_runtime.h>

// FlowStateCausalRevIN forward, mode='norm', x: [B=64, T=8192, C=64] f32.
// One wave32 owns one (batch, 16-channel group) and scans T sequentially.
// Intra-tile (16 time steps) inclusive prefix sums run on the matrix pipes:
// P = L @ X + carry*J with L = lower-triangular ones, chained as 4x
// v_wmma_f32_16x16x4_f32 (carry splatted into the C operand).
// Tiles are software-pipelined: tile s+1's global loads are issued before
// tile s is processed, hiding HBM/L2 latency behind the WMMA/VALU chain.

typedef __attribute__((ext_vector_type(2))) float v2f;
typedef __attribute__((ext_vector_type(8))) float v8f;

#define B_DIM 64
#define T_DIM 8192
#define C_DIM 64
#define NTILE (T_DIM / 16)
#define EPSV  1e-5f
#define PFD   8  /* prefetch distance in 16-step tiles */

// Inclusive 16-step prefix scan of one 16x16 tile: returns L*X + C.
// e[j] = row (j + 8*half) at this lane's channel; o[j] = row (j + 8*(1-half)).
// B-operand layout (mirrors documented A layout): VGPR0 = K(4k+2*half),
// VGPR1 = K+1.
__device__ __forceinline__ v8f scan16(const v2f A[4], int halfh,
                                      const float e[8], const float o[8],
                                      v8f c) {
  v2f Bv;
  Bv.x = halfh ? o[2] : e[0];
  Bv.y = halfh ? o[3] : e[1];
  c = __builtin_amdgcn_wmma_f32_16x16x4_f32(false, A[0], false, Bv, (short)0,
                                            c, false, false);
  Bv.x = halfh ? o[6] : e[4];
  Bv.y = halfh ? o[7] : e[5];
  c = __builtin_amdgcn_wmma_f32_16x16x4_f32(false, A[1], false, Bv, (short)0,
                                            c, false, false);
  Bv.x = halfh ? e[2] : o[0];
  Bv.y = halfh ? e[3] : o[1];
  c = __builtin_amdgcn_wmma_f32_16x16x4_f32(false, A[2], false, Bv, (short)0,
                                            c, false, false);
  Bv.x = halfh ? e[6] : o[4];
  Bv.y = halfh ? e[7] : o[5];
  c = __builtin_amdgcn_wmma_f32_16x16x4_f32(false, A[3], false, Bv, (short)0,
                                            c, false, false);
  return c;
}

// Issue both half-row register images of one 16x16 tile. The "o" rows are the
// same cache lines the partner half-wave pulls as "e" rows, so they hit WGP$;
// loading them directly (instead of ds_bpermute) lets them pipeline with VMEM.
__device__ __forceinline__ void load_tile(const float* __restrict__ tile,
                                          int halfh, float e[8], float o[8]) {
#pragma unroll
  for (int j = 0; j < 8; ++j)
    e[j] = tile[(size_t)(j + 8 * halfh) * C_DIM];
#pragma unroll
  for (int j = 0; j < 8; ++j)
    o[j] = tile[(size_t)(j + 8 * (1 - halfh)) * C_DIM];
}

__device__ __forceinline__ void process_tile(
    const float e[8], const float o[8], float* __restrict__ tout,
    const float* __restrict__ s_inv, int s, int halfh, const v2f Amat[4],
    float& carry_x, float& carry_d) {
  // First scan: causal sums of x. Carry rides in the C operand.
  v8f acc;
#pragma unroll
  for (int j = 0; j < 8; ++j) acc[j] = carry_x;
  acc = scan16(Amat, halfh, e, o, acc);

  float dd[8], d2[8], invn[8];
#pragma unroll
  for (int j = 0; j < 8; ++j) {
    const int t = s * 16 + j + 8 * halfh;
    const float inv = s_inv[t];  // same-address broadcast per half-wave
    invn[j] = inv;
    const float mean = acc[j] * inv;  // carry already included
    dd[j] = e[j] - mean;
    d2[j] = dd[j] * dd[j];
  }
  {  // new carry = inclusive sum through row 15 (upper half's acc[7])
    const float tot_o = __shfl_xor((float)acc[7], 16, 32);
    carry_x = halfh ? (float)acc[7] : tot_o;
  }

  // Second scan: causal sums of (x - mean)^2.
  float d2o[8];
#pragma unroll
  for (int j = 0; j < 8; ++j) d2o[j] = __shfl_xor(d2[j], 16, 32);
  v8f acc2;
#pragma unroll
  for (int j = 0; j < 8; ++j) acc2[j] = carry_d;
  acc2 = scan16(Amat, halfh, d2, d2o, acc2);

#pragma unroll
  for (int j = 0; j < 8; ++j) {
    const float var = acc2[j] * invn[j];
    const float rstd = rsqrtf(fmaxf(var, EPSV));
    tout[(size_t)(j + 8 * halfh) * C_DIM] = dd[j] * rstd;
  }
  {
    const float tot_o = __shfl_xor((float)acc2[7], 16, 32);
    carry_d = halfh ? (float)acc2[7] : tot_o;
  }
}

__global__ __launch_bounds__(128) void revin_causal_norm_kernel(
    const float* __restrict__ x, float* __restrict__ out) {
  // Exact 1/(t+1) table, built once per workgroup (32 KB LDS).
  __shared__ float s_inv[T_DIM];
  const int tid = threadIdx.x;
  for (int t = tid; t < T_DIM; t += 128) s_inv[t] = 1.0f / (float)(t + 1);
  __syncthreads();

  const int lane  = tid & 31;
  const int wave  = tid >> 5;        // channel group 0..3
  const int b     = blockIdx.x;      // batch
  const int n16   = lane & 15;       // channel within group == WMMA column
  const int halfh = (lane >> 4) & 1; // half-wave selector
  const int cbase = wave * 16;

  const float* __restrict__ xb = x + (size_t)b * T_DIM * C_DIM + cbase + n16;
  float* __restrict__ ob = out + (size_t)b * T_DIM * C_DIM + cbase + n16;

  // A = 16x16 lower-triangular ones, as four 16x4 column slices.
  // A layout: lanes hold M = lane&15; VGPR0 = K(4k+2*half), VGPR1 = K+1.
  v2f Amat[4];
#pragma unroll
  for (int k = 0; k < 4; ++k) {
    const int k0 = 4 * k + 2 * halfh;
    Amat[k].x = (k0 <= n16) ? 1.0f : 0.0f;
    Amat[k].y = (k0 + 1 <= n16) ? 1.0f : 0.0f;
  }

  float carry_x = 0.0f;  // per-channel running sum of x (incl. in C operand)
  float carry_d = 0.0f;  // per-channel running sum of (x-mean)^2

  // Software pipeline: two register tile buffers, unrolled by 2.
  float ea[8], oa[8], eb[8], ob_[8];
  load_tile(xb, halfh, ea, oa);

  for (int su = 0; su < NTILE / 2; ++su) {
    const int s0 = 2 * su;
    const int s1 = s0 + 1;
    const int s2 = (s1 + 1 < NTILE) ? s1 + 1 : NTILE - 1;  // clamped tail
    const int sp = (s0 + PFD < NTILE) ? s0 + PFD : NTILE - 1;

    // Issue next tile's loads, then compute current (latency overlap).
    load_tile(xb + (size_t)s1 * 16 * C_DIM, halfh, eb, ob_);
    __builtin_prefetch(xb + (size_t)sp * 16 * C_DIM, 0, 1);
    process_tile(ea, oa, ob + (size_t)s0 * 16 * C_DIM, s_inv, s0, halfh, Amat,
                 carry_x, carry_d);

    load_tile(xb + (size_t)s2 * 16 * C_DIM, halfh, ea, oa);
    process_tile(eb, ob_, ob + (size_t)s1 * 16 * C_DIM, s_inv, s1, halfh, Amat,
                 carry_x, carry_d);
  }
}

extern "C" void kernel_launch(void* const* d_in, const int* in_sizes, int n_in,
                              void* d_out, int out_size, void* d_ws,
                              size_t ws_size, hipStream_t stream) {
  (void)in_sizes; (void)n_in; (void)out_size; (void)d_ws; (void)ws_size;
  const float* x = (const float*)d_in[0];
  float* out = (float*)d_out;
  // 64 workgroups (one per batch) x 128 threads (4 waves = 4 channel groups).
  revin_causal_norm_kernel<<<dim3(B_DIM), dim3(128), 0, stream>>>(x, out);
}